// VQEmbeddingEMA_2018634629604
// MI455X (gfx1250) — compile-verified
//
#include <hip/hip_runtime.h>
#include <hip/hip_bf16.h>

typedef __attribute__((ext_vector_type(2))) float v2f;
typedef __attribute__((ext_vector_type(8))) float v8f;
typedef int v4i __attribute__((vector_size(16)));   // matches builtin param type

#define AS1 __attribute__((address_space(1)))
#define AS3 __attribute__((address_space(3)))

// Use gfx1250 async VMEM->LDS engine (ASYNCcnt) when the toolchain exposes it.
#if __has_builtin(__builtin_amdgcn_global_load_async_to_lds_b128) && \
    __has_builtin(__builtin_amdgcn_s_wait_asynccnt)
#define VQ_ASYNC_STAGE 1
#else
#define VQ_ASYNC_STAGE 0
#endif

// Problem constants (match reference)
constexpr int Dn = 256;       // feature dim
constexpr int Mn = 1024;      // codebook size
constexpr int Nn = 32768;     // B*H*W vectors
constexpr float DECAY = 0.999f;
constexpr float ONE_MINUS_DECAY = 0.001f;
constexpr float EPS_C = 1e-5f;

// Output layout (flat, in reference return order)
constexpr size_t OUT_Q      = 0;                       // N*D
constexpr size_t OUT_COMMIT = 8388608;                 // 1
constexpr size_t OUT_CODE   = 8388609;                 // 1
constexpr size_t OUT_PERP   = 8388610;                 // 1
constexpr size_t OUT_EMB    = 8388611;                 // M*D
constexpr size_t OUT_CNT    = 8650755;                 // M
constexpr size_t OUT_W      = 8651779;                 // M*D

// ---------------------------------------------------------------------------
// Kernel 1: e_sq[m] = sum_k E[m,k]^2
// ---------------------------------------------------------------------------
__global__ __launch_bounds__(256) void vq_esq(const float* __restrict__ emb,
                                              float* __restrict__ esq) {
  const int m = blockIdx.x * 256 + threadIdx.x;   // grid = 4 -> 1024
  const float* p = emb + (size_t)m * Dn;
  float s = 0.f;
  #pragma unroll 8
  for (int k = 0; k < Dn; k += 4) {
    float4 v = *(const float4*)(p + k);
    s += v.x * v.x + v.y * v.y + v.z * v.z + v.w * v.w;
  }
  esq[m] = s;
}

// ---------------------------------------------------------------------------
// Kernel 2: fused distance + argmin using V_WMMA_F32_16X16X4_F32.
// 256 blocks x 256 threads (8 waves). Wave w owns 16 rows of x; block stages
// 16 embedding rows (one col-chunk) into LDS, all waves consume it.
// Staging uses double-buffered GLOBAL_LOAD_ASYNC_TO_LDS_B128 (ASYNCcnt) so
// the copy of chunk k+1 overlaps the 64 WMMAs of chunk k.
// dist_proxy = e_sq[j] - 2 * <x_i, e_j>   (x_sq dropped: constant per row)
// ---------------------------------------------------------------------------
__global__ __launch_bounds__(256) void vq_argmin_wmma(
    const float* __restrict__ x, const float* __restrict__ emb,
    const float* __restrict__ esq, int* __restrict__ indices) {
#if VQ_ASYNC_STAGE
  __shared__ float ldsE[2][16 * Dn];              // 2 x 16 KB double buffer
#else
  __shared__ float ldsE[1][16 * Dn];              // 16 KB single buffer
#endif
  const int t    = threadIdx.x;
  const int lane = t & 31;
  const int wave = t >> 5;                        // 0..7
  const int lr   = lane & 15;                     // A: row-in-tile, B: col-in-chunk
  const int hi   = lane >> 4;                     // K-pair select
  const int rowBase = blockIdx.x * 128 + wave * 16;

  // Staging geometry: thread t moves 64 B of chunk-row (t>>4).
  const int sc   = t >> 4;
  const int kseg = (t & 15) * 16;

  // A fragments for this wave's 16 rows over full K=256 (64 steps of K=4).
  // Lane layout: afrag[s] = x[rowBase+lr, 4*s + 2*hi + {0,1}]
  v2f afrag[64];
  {
    const float* xr = x + (size_t)(rowBase + lr) * Dn + 2 * hi;
    #pragma unroll
    for (int s = 0; s < 64; ++s) afrag[s] = *(const v2f*)(xr + 4 * s);
  }

  float bestVal[8];
  int   bestIdx[8];
  #pragma unroll
  for (int r = 0; r < 8; ++r) { bestVal[r] = 3.4e38f; bestIdx[r] = 0; }

#if VQ_ASYNC_STAGE
  // Issue async copy of chunk 0 into buffer 0.
  {
    const float* src = emb + (size_t)(0 * 16 + sc) * Dn + kseg;
    float* dst = &ldsE[0][sc * Dn + kseg];
    #pragma unroll
    for (int i = 0; i < 4; ++i)
      __builtin_amdgcn_global_load_async_to_lds_b128(
          (AS1 v4i*)(src + 4 * i), (AS3 v4i*)(dst + 4 * i), 0, 0);
  }
#endif

  for (int chunk = 0; chunk < 64; ++chunk) {
#if VQ_ASYNC_STAGE
    const int buf = chunk & 1;
    __builtin_amdgcn_s_wait_asynccnt(0);          // my async copies landed
    __syncthreads();                              // everyone's copies visible
    if (chunk < 63) {                             // overlap next chunk's copy
      const float* src = emb + (size_t)((chunk + 1) * 16 + sc) * Dn + kseg;
      float* dst = &ldsE[buf ^ 1][sc * Dn + kseg];
      #pragma unroll
      for (int i = 0; i < 4; ++i)
        __builtin_amdgcn_global_load_async_to_lds_b128(
            (AS1 v4i*)(src + 4 * i), (AS3 v4i*)(dst + 4 * i), 0, 0);
    }
#else
    const int buf = 0;
    {
      const float4* src = (const float4*)(emb + (size_t)(chunk * 16 + sc) * Dn + kseg);
      float4* dst = (float4*)(&ldsE[0][sc * Dn + kseg]);
      dst[0] = src[0]; dst[1] = src[1]; dst[2] = src[2]; dst[3] = src[3];
      if (chunk < 63)  // global_prefetch_b8 of next chunk
        __builtin_prefetch(emb + (size_t)((chunk + 1) * 16 + sc) * Dn + kseg, 0, 1);
    }
    __syncthreads();
#endif

    // 16x16 score tile: 64 x V_WMMA_F32_16X16X4_F32 over K=256.
    v8f acc = {};
    const float* Bp = &ldsE[buf][lr * Dn + 2 * hi]; // B: col = lr, K-pair by hi
    #pragma unroll
    for (int s = 0; s < 64; ++s) {
      v2f b = *(const v2f*)(Bp + 4 * s);
      acc = __builtin_amdgcn_wmma_f32_16x16x4_f32(
          false, afrag[s], false, b, (short)0, acc, false, false);
    }

    // C layout: element r -> row = r + 8*hi, col = chunk*16 + lr.
    const int col = chunk * 16 + lr;
    const float e2 = esq[col];
    #pragma unroll
    for (int r = 0; r < 8; ++r) {
      float dist = e2 - 2.0f * acc[r];
      if (dist < bestVal[r]) { bestVal[r] = dist; bestIdx[r] = col; }
    }
#if !VQ_ASYNC_STAGE
    __syncthreads();                              // buffer reused next chunk
#endif
  }

  // Argmin across the 16 lanes of each half-wave (cols), lowest-index ties.
  #pragma unroll
  for (int r = 0; r < 8; ++r) {
    #pragma unroll
    for (int off = 8; off >= 1; off >>= 1) {
      float ov = __shfl_xor(bestVal[r], off, 16);
      int   oi = __shfl_xor(bestIdx[r], off, 16);
      if (ov < bestVal[r] || (ov == bestVal[r] && oi < bestIdx[r])) {
        bestVal[r] = ov; bestIdx[r] = oi;
      }
    }
    if (lr == 0) indices[rowBase + r + 8 * hi] = bestIdx[r];
  }
}

// ---------------------------------------------------------------------------
// Kernel 3: per-code deterministic scan (replaces scatter-add):
//   dw[m,:] = sum_{n: idx[n]==m} x[n,:]   (fixed n order -> deterministic)
//   new_weight = DECAY*ema_weight + (1-DECAY)*dw ; counts[m] = #matches
// ---------------------------------------------------------------------------
__global__ __launch_bounds__(256) void vq_ema_weight(
    const int* __restrict__ indices, const float* __restrict__ x,
    const float* __restrict__ ema_weight, float* __restrict__ out_weight,
    float* __restrict__ counts) {
  const int m = blockIdx.x;        // 0..1023
  const int t = threadIdx.x;       // feature col 0..255
  __shared__ int sIdx[256];
  float acc = 0.f;
  int cnt = 0;
  for (int base = 0; base < Nn; base += 256) {
    __syncthreads();
    sIdx[t] = indices[base + t];
    __syncthreads();
    for (int j = 0; j < 256; ++j) {
      if (sIdx[j] == m) {          // uniform branch across the block
        acc += x[(size_t)(base + j) * Dn + t];
        ++cnt;
      }
    }
  }
  out_weight[(size_t)m * Dn + t] =
      DECAY * ema_weight[(size_t)m * Dn + t] + ONE_MINUS_DECAY * acc;
  if (t == 0) counts[m] = (float)cnt;
}

// ---------------------------------------------------------------------------
// Kernel 4: gather quantized (== quantized_st numerically) + sq-error partials
// ---------------------------------------------------------------------------
__global__ __launch_bounds__(256) void vq_quantize_loss(
    const int* __restrict__ indices, const float* __restrict__ x,
    const float* __restrict__ emb, float* __restrict__ out_q,
    float* __restrict__ partial) {
  const int t = threadIdx.x;
  const int row0 = blockIdx.x * 128;   // grid = 256
  float local = 0.f;
  for (int i = 0; i < 128; ++i) {
    const int row = row0 + i;
    const int idx = indices[row];
    const float q  = emb[(size_t)idx * Dn + t];
    const float xv = x[(size_t)row * Dn + t];
    out_q[(size_t)row * Dn + t] = q;
    const float d = xv - q;
    local += d * d;
  }
  __shared__ float red[256];
  red[t] = local;
  __syncthreads();
  for (int off = 128; off >= 1; off >>= 1) {
    if (t < off) red[t] += red[t + off];
    __syncthreads();
  }
  if (t == 0) partial[blockIdx.x] = red[0];
}

// ---------------------------------------------------------------------------
// Kernel 5: finalize (1 block x 1024): new_count (Laplace-normalized),
// perplexity, commitment/codebook loss.
// ---------------------------------------------------------------------------
__global__ __launch_bounds__(1024) void vq_finalize(
    const float* __restrict__ counts, const float* __restrict__ ema_count,
    const float* __restrict__ partial, float* __restrict__ out) {
  const int m = threadIdx.x;
  __shared__ float red[1024];

  const float raw = DECAY * ema_count[m] + ONE_MINUS_DECAY * counts[m];
  red[m] = raw;
  __syncthreads();
  for (int off = 512; off >= 1; off >>= 1) {
    if (m < off) red[m] += red[m + off];
    __syncthreads();
  }
  const float nsum = red[0];
  __syncthreads();
  out[OUT_CNT + m] = (raw + EPS_C) / (nsum + (float)Mn * EPS_C) * nsum;

  const float p = counts[m] * (1.0f / (float)Nn);
  red[m] = -p * logf(p + 1e-10f);
  __syncthreads();
  for (int off = 512; off >= 1; off >>= 1) {
    if (m < off) red[m] += red[m + off];
    __syncthreads();
  }
  if (m == 0) out[OUT_PERP] = expf(red[0]);
  __syncthreads();

  red[m] = (m < 256) ? partial[m] : 0.f;
  __syncthreads();
  for (int off = 512; off >= 1; off >>= 1) {
    if (m < off) red[m] += red[m + off];
    __syncthreads();
  }
  if (m == 0) {
    const float mse = red[0] / ((float)Nn * (float)Dn);
    out[OUT_CODE]   = mse;          // codebook_loss
    out[OUT_COMMIT] = 0.25f * mse;  // commitment_loss (same mean numerically)
  }
}

// ---------------------------------------------------------------------------
// Kernel 6: new_embedding = new_weight / new_count[:,None]
// ---------------------------------------------------------------------------
__global__ __launch_bounds__(256) void vq_new_embedding(
    const float* __restrict__ out_w, const float* __restrict__ out_cnt,
    float* __restrict__ out_emb) {
  const int m = blockIdx.x;        // 1024
  const int t = threadIdx.x;       // 256
  out_emb[(size_t)m * Dn + t] = out_w[(size_t)m * Dn + t] / out_cnt[m];
}

// ---------------------------------------------------------------------------
extern "C" void kernel_launch(void* const* d_in, const int* in_sizes, int n_in,
                              void* d_out, int out_size, void* d_ws, size_t ws_size,
                              hipStream_t stream) {
  const float* x          = (const float*)d_in[0];   // [N, D]
  const float* emb        = (const float*)d_in[1];   // [M, D]
  const float* ema_count  = (const float*)d_in[2];   // [M]
  const float* ema_weight = (const float*)d_in[3];   // [M, D]
  float* out = (float*)d_out;

  // Workspace layout (~140 KB)
  int*   ws_idx     = (int*)d_ws;                              // N ints
  float* ws_esq     = (float*)((char*)d_ws + 131072);          // M floats
  float* ws_cnt     = ws_esq + Mn;                             // M floats
  float* ws_partial = ws_cnt + Mn;                             // 256 floats

  float* out_q   = out + OUT_Q;
  float* out_emb = out + OUT_EMB;
  float* out_cnt = out + OUT_CNT;
  float* out_w   = out + OUT_W;

  vq_esq<<<Mn / 256, 256, 0, stream>>>(emb, ws_esq);
  vq_argmin_wmma<<<Nn / 128, 256, 0, stream>>>(x, emb, ws_esq, ws_idx);
  vq_ema_weight<<<Mn, 256, 0, stream>>>(ws_idx, x, ema_weight, out_w, ws_cnt);
  vq_quantize_loss<<<Nn / 128, 256, 0, stream>>>(ws_idx, x, emb, out_q, ws_partial);
  vq_finalize<<<1, 1024, 0, stream>>>(ws_cnt, ema_count, ws_partial, out);
  vq_new_embedding<<<Mn, 256, 0, stream>>>(out_w, out_cnt, out_emb);
}